// SparseMaxPool2dTestTorch_16681652977715
// MI455X (gfx1250) — compile-verified
//
#include <hip/hip_runtime.h>
#include <math.h>

// Problem constants from the reference: B=4, H=512, W=512, C=32, K=3, S=2, P=1
#define HO 256
#define WO 256
#define CC 32

// CDNA5 device-scope non-returning float atomic max (IEEE maximumNumber).
// ISA 15.18.3 opcode 82: GLOBAL_ATOMIC_MAX_NUM_F32. Default TH[0]=0 => no
// return value (STOREcnt-tracked). scope:SCOPE_DEV makes the RMW coherent
// across all WGPs; the 32MB output stays resident in the 192MB L2, so these
// atomics are serviced by the L2 atomic units without HBM round-trips.
__device__ __forceinline__ void global_atomic_max_f32_dev(float* addr, float val) {
    asm volatile("global_atomic_max_num_f32 %0, %1, off scope:SCOPE_DEV"
                 :
                 : "v"(addr), "v"(val)
                 : "memory");
}

// Pass 1: fill output with -inf (identity of max). 128-bit stores.
__global__ void init_neginf_kernel(float4* __restrict__ out, int n4) {
    int i = blockIdx.x * blockDim.x + threadIdx.x;
    if (i < n4) {
        const float ninf = -__builtin_inff();
        float4 v;
        v.x = ninf; v.y = ninf; v.z = ninf; v.w = ninf;
        out[i] = v;
    }
}

// Pass 2: scatter-max. One wave32 per point, lane == channel (C==32 matches
// CDNA5 wave32 exactly): the wave reads one 128B feature row fully coalesced.
// For K=3,S=2,P=1 each input pixel (y,x) contributes to output rows
// {y>>1,(y+1)>>1} x cols {x>>1,(x+1)>>1} -> 1..4 cells (avg 2.25).
__global__ void scatter_max_kernel(const float* __restrict__ feats,
                                   const int*   __restrict__ coors,
                                   float*       __restrict__ out,
                                   int npts) {
    int t    = blockIdx.x * blockDim.x + threadIdx.x;
    int n    = t >> 5;        // point index (one wave32 per point)
    int lane = t & 31;        // channel index == lane id
    if (n >= npts) return;

    // n is wave-uniform by construction; readfirstlane lets the compiler
    // prove it and turn the 3 coordinate loads into scalar (s_load) fetches
    // through the constant cache instead of 3 per-lane VMEM loads.
    int nu = __builtin_amdgcn_readfirstlane(n);
    int b  = coors[3 * nu + 0];
    int y  = coors[3 * nu + 1];
    int x  = coors[3 * nu + 2];

    float f = feats[n * CC + lane];   // 128B coalesced per wave

    int oy0 = y >> 1;
    int oy1 = (y + 1) >> 1;       // == oy0 for even y
    int ox0 = x >> 1;
    int ox1 = (x + 1) >> 1;       // == ox0 for even x
    if (oy1 >= HO) oy1 = oy0;     // y == 511 edge
    if (ox1 >= WO) ox1 = ox0;     // x == 511 edge

    // 32MB output -> all offsets fit in 32 bits; avoid 64-bit VALU chains.
    int base = b * (HO * WO * CC) + lane;
    float* o = out;

    global_atomic_max_f32_dev(o + base + (oy0 * WO + ox0) * CC, f);
    if (ox1 != ox0)
        global_atomic_max_f32_dev(o + base + (oy0 * WO + ox1) * CC, f);
    if (oy1 != oy0) {
        global_atomic_max_f32_dev(o + base + (oy1 * WO + ox0) * CC, f);
        if (ox1 != ox0)
            global_atomic_max_f32_dev(o + base + (oy1 * WO + ox1) * CC, f);
    }
}

// Pass 3: empty cells stayed -inf -> 0 (reference: where(isinf(out), 0, out)).
__global__ void finalize_kernel(float4* __restrict__ out, int n4) {
    int i = blockIdx.x * blockDim.x + threadIdx.x;
    if (i < n4) {
        float4 v = out[i];
        v.x = __builtin_isinf(v.x) ? 0.0f : v.x;
        v.y = __builtin_isinf(v.y) ? 0.0f : v.y;
        v.z = __builtin_isinf(v.z) ? 0.0f : v.z;
        v.w = __builtin_isinf(v.w) ? 0.0f : v.w;
        out[i] = v;
    }
}

extern "C" void kernel_launch(void* const* d_in, const int* in_sizes, int n_in,
                              void* d_out, int out_size, void* d_ws, size_t ws_size,
                              hipStream_t stream) {
    (void)n_in; (void)d_ws; (void)ws_size;

    const float* feats = (const float*)d_in[0];   // (N, 32) fp32
    const int*   coors = (const int*)d_in[1];     // (N, 3)  int32
    float*       out   = (float*)d_out;           // (4, 256, 256, 32) fp32

    const int npts = in_sizes[0] / CC;            // N = 300000
    const int n4   = out_size / 4;                // 2,097,152 float4

    const int T = 256;                            // 8 waves / block

    // Pass 1: init output to -inf
    init_neginf_kernel<<<(n4 + T - 1) / T, T, 0, stream>>>((float4*)out, n4);

    // Pass 2: one wave32 per point
    long long threads = (long long)npts * 32;
    int blocks = (int)((threads + T - 1) / T);
    scatter_max_kernel<<<blocks, T, 0, stream>>>(feats, coors, out, npts);

    // Pass 3: -inf -> 0
    finalize_kernel<<<(n4 + T - 1) / T, T, 0, stream>>>((float4*)out, n4);
}